// GCNEncoder_45131516346603
// MI455X (gfx1250) — compile-verified
//
#include <hip/hip_runtime.h>
#include <math.h>

#define N_NODES 100000
#define N_EDGES 800000
#define B_GRAPHS 64
#define IN_F 64
#define H_F 256
#define OUT_F 128
#define EPS_BN 1e-5f
#define EPS_NORM 1e-12f

typedef float v2f __attribute__((ext_vector_type(2)));
typedef float v8f __attribute__((ext_vector_type(8)));

// ---------------- utility ----------------
__global__ void zero_kernel(float* __restrict__ p, int n) {
    int i = blockIdx.x * blockDim.x + threadIdx.x;
    if (i < n) p[i] = 0.0f;
}

// ---------------- degree / normalization ----------------
__global__ void deg_init_kernel(float* __restrict__ deg) {
    int i = blockIdx.x * blockDim.x + threadIdx.x;
    if (i < N_NODES) deg[i] = 1.0f;   // self loop
}

__global__ void deg_edge_kernel(const long long* __restrict__ ei, float* __restrict__ deg) {
    int e = blockIdx.x * blockDim.x + threadIdx.x;
    if (e < N_EDGES) {
        int dst = (int)ei[N_EDGES + e];      // edge_index[1] = targets
        atomicAdd(&deg[dst], 1.0f);
    }
}

__global__ void dinv_kernel(float* __restrict__ deg) {
    int i = blockIdx.x * blockDim.x + threadIdx.x;
    if (i < N_NODES) {
        float d = deg[i];
        deg[i] = (d > 0.0f) ? rsqrtf(d) : 0.0f;
    }
}

// ---------------- register-blocked WMMA GEMM ----------------
// Y[N x MOUT] = X[N x K] @ W[K x MOUT], fp32 via v_wmma_f32_16x16x4_f32.
// One wave computes a (16*RT) x (16*CT) macro-tile: per K-step of 4 it loads
// RT A-pairs + CT B-pairs (b64 each) and issues RT*CT WMMAs -> ~16 FLOP/B.
// A 16x4 layout: lane l -> row m=l&15; VGPR0/1 hold K = 2*(l>>4) + {0,1}.
// B 4x16 layout: lane l -> col n=l&15; VGPR0/1 hold K = 2*(l>>4) + {0,1}.
// C/D 16x16:     lane l -> col n=l&15; VGPR v holds row m = v + 8*(l>>4).
template <int K, int MOUT, int RT, int CT>
__global__ __launch_bounds__(256) void gemm_wmma_f32_blk(const float* __restrict__ X,
                                                         const float* __restrict__ W,
                                                         float* __restrict__ Y) {
    const int lane = threadIdx.x & 31;
    const int wave = blockIdx.x * (blockDim.x >> 5) + (threadIdx.x >> 5);
    const int rowTiles  = N_NODES / 16;                 // 6250 (exact)
    const int rowGroups = (rowTiles + RT - 1) / RT;
    const int colGroups = MOUT / (16 * CT);
    const int rg = wave / colGroups;
    const int cg = wave % colGroups;
    if (rg >= rowGroups) return;

    const int m  = lane & 15;
    const int kb = (lane >> 4) << 1;                    // 0 or 2
    const int n0 = cg * (16 * CT) + (lane & 15);

    const float* xr[RT];
    bool tile_valid[RT];
#pragma unroll
    for (int rt = 0; rt < RT; ++rt) {
        const int tile = rg * RT + rt;
        tile_valid[rt] = (tile < rowTiles);             // tiles are all-or-nothing
        int row = tile * 16 + m;
        if (row >= N_NODES) row = N_NODES - 1;          // clamp: safe dummy reads
        xr[rt] = X + (size_t)row * K;
    }

    v8f acc[RT][CT] = {};
    for (int k0 = 0; k0 < K; k0 += 4) {
        v2f a[RT], b[CT];
#pragma unroll
        for (int rt = 0; rt < RT; ++rt) {
            a[rt].x = xr[rt][k0 + kb];
            a[rt].y = xr[rt][k0 + kb + 1];
        }
#pragma unroll
        for (int ct = 0; ct < CT; ++ct) {
            b[ct].x = W[(size_t)(k0 + kb) * MOUT + n0 + 16 * ct];
            b[ct].y = W[(size_t)(k0 + kb + 1) * MOUT + n0 + 16 * ct];
        }
#pragma unroll
        for (int rt = 0; rt < RT; ++rt)
#pragma unroll
            for (int ct = 0; ct < CT; ++ct)
                acc[rt][ct] = __builtin_amdgcn_wmma_f32_16x16x4_f32(
                    false, a[rt], false, b[ct], (short)0, acc[rt][ct], false, false);
    }

    const int half = (lane >> 4) << 3;
#pragma unroll
    for (int rt = 0; rt < RT; ++rt) {
        if (!tile_valid[rt]) continue;
        const int mbase = (rg * RT + rt) * 16 + half;
#pragma unroll
        for (int ct = 0; ct < CT; ++ct) {
            const int n = n0 + 16 * ct;
#pragma unroll
            for (int v = 0; v < 8; ++v)
                Y[(size_t)(mbase + v) * MOUT + n] = acc[rt][ct][v];
        }
    }
}

// ---------------- aggregation: agg = bias + dinv^2 * h (self loop) ----------------
template <int F>
__global__ void agg_init_kernel(const float* __restrict__ h, const float* __restrict__ dinv,
                                const float* __restrict__ bias, float* __restrict__ agg) {
    size_t idx = (size_t)blockIdx.x * blockDim.x + threadIdx.x;
    if (idx >= (size_t)N_NODES * F) return;
    int node = (int)(idx / F);
    int c    = (int)(idx % F);
    float di = dinv[node];
    agg[idx] = di * di * h[idx] + bias[c];
}

// ---------------- aggregation: scatter-add normalized messages over edges ----------------
template <int F>
__global__ void scatter_kernel(const float* __restrict__ h, const long long* __restrict__ ei,
                               const float* __restrict__ dinv, float* __restrict__ agg) {
    const int CPE = F / 4;                                  // float4 chunks per edge
    size_t idx = (size_t)blockIdx.x * blockDim.x + threadIdx.x;
    if (idx >= (size_t)N_EDGES * CPE) return;
    int e = (int)(idx / CPE);
    int c = (int)(idx % CPE);
    int src = (int)ei[e];                 // row (gather)
    int dst = (int)ei[N_EDGES + e];       // col (scatter)
    float nrm = dinv[src] * dinv[dst];
    const float4 v = ((const float4*)(h + (size_t)src * F))[c];
    float* o = agg + (size_t)dst * F + (size_t)c * 4;
    atomicAdd(o + 0, v.x * nrm);
    atomicAdd(o + 1, v.y * nrm);
    atomicAdd(o + 2, v.z * nrm);
    atomicAdd(o + 3, v.w * nrm);
}

// ---------------- batchnorm: stats (sum, sumsq per channel) ----------------
template <int F>
__global__ void bn_stats_kernel(const float* __restrict__ h, float* __restrict__ stats) {
    const int c = threadIdx.x;                // blockDim.x == F (coalesced across channels)
    int r0 = blockIdx.x * 512;
    int r1 = r0 + 512;
    if (r1 > N_NODES) r1 = N_NODES;
    float s = 0.0f, sq = 0.0f;
    for (int r = r0; r < r1; ++r) {
        float v = h[(size_t)r * F + c];
        s += v;
        sq += v * v;
    }
    atomicAdd(&stats[c], s);
    atomicAdd(&stats[F + c], sq);
}

// ---------------- batchnorm apply + ReLU (in place) ----------------
template <int F>
__global__ void bn_apply_kernel(float* __restrict__ h, const float* __restrict__ stats,
                                const float* __restrict__ g, const float* __restrict__ be) {
    size_t idx = (size_t)blockIdx.x * blockDim.x + threadIdx.x;
    if (idx >= (size_t)N_NODES * F) return;
    int c = (int)(idx % F);
    const float invN = 1.0f / (float)N_NODES;
    float mean = stats[c] * invN;
    float var  = stats[F + c] * invN - mean * mean;   // biased variance
    float inv  = rsqrtf(var + EPS_BN);
    float v = (h[idx] - mean) * inv * g[c] + be[c];
    h[idx] = fmaxf(v, 0.0f);
}

// ---------------- global mean pool (sums + counts via atomics) ----------------
__global__ void pool_kernel(const float* __restrict__ h, const long long* __restrict__ batch,
                            float* __restrict__ sums, float* __restrict__ cnts) {
    size_t idx = (size_t)blockIdx.x * blockDim.x + threadIdx.x;
    if (idx >= (size_t)N_NODES * OUT_F) return;
    int node = (int)(idx / OUT_F);
    int c    = (int)(idx % OUT_F);
    int b    = (int)batch[node];
    atomicAdd(&sums[b * OUT_F + c], h[idx]);
    if (c == 0) atomicAdd(&cnts[b], 1.0f);
}

// ---------------- finalize: mean + L2 row normalize ----------------
__global__ __launch_bounds__(OUT_F) void finalize_kernel(const float* __restrict__ sums,
                                                         const float* __restrict__ cnts,
                                                         float* __restrict__ out) {
    __shared__ float red[OUT_F];
    int g = blockIdx.x, t = threadIdx.x;
    float m = sums[g * OUT_F + t] / fmaxf(cnts[g], 1.0f);
    red[t] = m * m;
    __syncthreads();
    for (int s = OUT_F / 2; s > 0; s >>= 1) {
        if (t < s) red[t] += red[t + s];
        __syncthreads();
    }
    float nrm = fmaxf(sqrtf(red[0]), EPS_NORM);
    out[g * OUT_F + t] = m / nrm;
}

// ---------------- launcher ----------------
extern "C" void kernel_launch(void* const* d_in, const int* in_sizes, int n_in,
                              void* d_out, int out_size, void* d_ws, size_t ws_size,
                              hipStream_t stream) {
    const float*     x     = (const float*)d_in[0];
    const long long* ei    = (const long long*)d_in[1];
    const long long* batch = (const long long*)d_in[2];
    const float* W1 = (const float*)d_in[3];
    const float* b1 = (const float*)d_in[4];
    const float* g1 = (const float*)d_in[5];
    const float* be1 = (const float*)d_in[6];
    const float* W2 = (const float*)d_in[7];
    const float* b2 = (const float*)d_in[8];
    const float* g2 = (const float*)d_in[9];
    const float* be2 = (const float*)d_in[10];
    const float* W3 = (const float*)d_in[11];
    const float* b3 = (const float*)d_in[12];
    const float* g3 = (const float*)d_in[13];
    const float* be3 = (const float*)d_in[14];
    float* out = (float*)d_out;

    float* ws    = (float*)d_ws;
    float* bufA  = ws;                                   // N * H_F
    float* bufB  = bufA + (size_t)N_NODES * H_F;         // N * H_F
    float* dinv  = bufB + (size_t)N_NODES * H_F;         // N
    float* stats = dinv + N_NODES;                       // 2 * H_F
    float* psum  = stats + 2 * H_F;                      // B * OUT_F
    float* pcnt  = psum + B_GRAPHS * OUT_F;              // B

    const int T = 256;
    // degree -> dinv (in place)
    deg_init_kernel<<<(N_NODES + T - 1) / T, T, 0, stream>>>(dinv);
    deg_edge_kernel<<<(N_EDGES + T - 1) / T, T, 0, stream>>>(ei, dinv);
    dinv_kernel<<<(N_NODES + T - 1) / T, T, 0, stream>>>(dinv);

    const int gridNH  = (int)(((size_t)N_NODES * H_F + T - 1) / T);
    const int gridNO  = (int)(((size_t)N_NODES * OUT_F + T - 1) / T);
    const int gridEH  = (int)(((size_t)N_EDGES * (H_F / 4) + T - 1) / T);
    const int gridEO  = (int)(((size_t)N_EDGES * (OUT_F / 4) + T - 1) / T);
    const int rowBlks = (N_NODES + 511) / 512;

    // GEMM grids: waves = rowGroups * colGroups, 8 waves per 256-thread block
    const int rowTiles  = N_NODES / 16;                  // 6250
    const int rowGroups = (rowTiles + 3) / 4;            // 1563 (RT = 4)
    const int gridGemmH = (rowGroups * (H_F / 64) + 7) / 8;    // CT=4 -> 4 col groups
    const int gridGemmO = (rowGroups * (OUT_F / 64) + 7) / 8;  // CT=4 -> 2 col groups

    // ---- layer 1: x (N x 64) -> bufA -> agg bufB -> bn/relu in place ----
    gemm_wmma_f32_blk<IN_F, H_F, 4, 4><<<gridGemmH, T, 0, stream>>>(x, W1, bufA);
    agg_init_kernel<H_F><<<gridNH, T, 0, stream>>>(bufA, dinv, b1, bufB);
    scatter_kernel<H_F><<<gridEH, T, 0, stream>>>(bufA, ei, dinv, bufB);
    zero_kernel<<<(2 * H_F + T - 1) / T, T, 0, stream>>>(stats, 2 * H_F);
    bn_stats_kernel<H_F><<<rowBlks, H_F, 0, stream>>>(bufB, stats);
    bn_apply_kernel<H_F><<<gridNH, T, 0, stream>>>(bufB, stats, g1, be1);

    // ---- layer 2: bufB (N x 256) -> bufA -> agg bufB -> bn/relu ----
    gemm_wmma_f32_blk<H_F, H_F, 4, 4><<<gridGemmH, T, 0, stream>>>(bufB, W2, bufA);
    agg_init_kernel<H_F><<<gridNH, T, 0, stream>>>(bufA, dinv, b2, bufB);
    scatter_kernel<H_F><<<gridEH, T, 0, stream>>>(bufA, ei, dinv, bufB);
    zero_kernel<<<(2 * H_F + T - 1) / T, T, 0, stream>>>(stats, 2 * H_F);
    bn_stats_kernel<H_F><<<rowBlks, H_F, 0, stream>>>(bufB, stats);
    bn_apply_kernel<H_F><<<gridNH, T, 0, stream>>>(bufB, stats, g2, be2);

    // ---- layer 3: bufB (N x 256) -> bufA (N x 128) -> agg bufB -> bn/relu ----
    gemm_wmma_f32_blk<H_F, OUT_F, 4, 4><<<gridGemmO, T, 0, stream>>>(bufB, W3, bufA);
    agg_init_kernel<OUT_F><<<gridNO, T, 0, stream>>>(bufA, dinv, b3, bufB);
    scatter_kernel<OUT_F><<<gridEO, T, 0, stream>>>(bufA, ei, dinv, bufB);
    zero_kernel<<<(2 * H_F + T - 1) / T, T, 0, stream>>>(stats, 2 * H_F);
    bn_stats_kernel<OUT_F><<<rowBlks, OUT_F, 0, stream>>>(bufB, stats);
    bn_apply_kernel<OUT_F><<<gridNO, T, 0, stream>>>(bufB, stats, g3, be3);

    // ---- pool + normalize ----
    zero_kernel<<<(B_GRAPHS * OUT_F + B_GRAPHS + T - 1) / T, T, 0, stream>>>(psum,
                                                               B_GRAPHS * OUT_F + B_GRAPHS);
    pool_kernel<<<gridNO, T, 0, stream>>>(bufB, batch, psum, pcnt);
    finalize_kernel<<<B_GRAPHS, OUT_F, 0, stream>>>(psum, pcnt, out);
}